// CondNNWAvgPRModel_76227079569955
// MI455X (gfx1250) — compile-verified
//
#include <hip/hip_runtime.h>
#include <math.h>

#define B_ 256
#define N_ 1024
#define D_ 128
#define ND_ (D_*D_)            // 16384
#define NSPLIT 4
#define ROWS_PER_SPLIT (N_/NSPLIT)   // 256

typedef __attribute__((ext_vector_type(2))) float v2f;
typedef __attribute__((ext_vector_type(8))) float v8f;

// ---------------------------------------------------------------------------
// Kernel 1: mask-weighted reductions over N (pure bandwidth, 402 MB stream).
//   Dmd[b][p] = sum_n (mi-mj) * d[b,n,p]
//   Sd [b][p] = sum_n mi*si[b,n,p] - mj*sj[b,n,p]
//   cnt[b]    = sum_n (mi-mj)
// Grid: B*NSPLIT blocks of 256 threads; each block does one (batch, N-chunk).
// Per-split slabs -> no atomics, no zero-init required.
// ---------------------------------------------------------------------------
__global__ __launch_bounds__(256) void k1_reduce(
    const float* __restrict__ d, const float* __restrict__ si, const float* __restrict__ sj,
    const unsigned char* __restrict__ mi, const unsigned char* __restrict__ mj,
    float* __restrict__ wsDm, float* __restrict__ wsSd,
    float* __restrict__ wsCnt, float* __restrict__ acc)
{
    const int bid = blockIdx.x;
    const int b   = bid >> 2;        // batch
    const int s   = bid & 3;         // N-split
    const int t   = threadIdx.x;
    const int cg  = t & 31;          // column group: 4 floats each (128 cols / 32)
    const int rg  = t >> 5;          // row group (8 waves)

    float4 accD = make_float4(0.f, 0.f, 0.f, 0.f);
    float4 accS = make_float4(0.f, 0.f, 0.f, 0.f);
    float  cnt  = 0.f;

    const long row0 = (long)b * N_ + (long)s * ROWS_PER_SPLIT;
    #pragma unroll 4
    for (int i = 0; i < ROWS_PER_SPLIT / 8; ++i) {
        const long row = row0 + rg + i * 8;
        const float wi = (float)mi[row];
        const float wj = (float)mj[row];
        const float dw = wi - wj;
        const float4 dv  = ((const float4*)(d  + row * D_))[cg];
        const float4 siv = ((const float4*)(si + row * D_))[cg];
        const float4 sjv = ((const float4*)(sj + row * D_))[cg];
        accD.x += dw * dv.x;  accD.y += dw * dv.y;
        accD.z += dw * dv.z;  accD.w += dw * dv.w;
        accS.x += wi * siv.x - wj * sjv.x;
        accS.y += wi * siv.y - wj * sjv.y;
        accS.z += wi * siv.z - wj * sjv.z;
        accS.w += wi * siv.w - wj * sjv.w;
        if (cg == 0) cnt += dw;      // count once per row
    }

    __shared__ float ldsD[8][128];
    __shared__ float ldsS[8][128];
    __shared__ float ldsC[256];
    ldsD[rg][cg*4+0] = accD.x; ldsD[rg][cg*4+1] = accD.y;
    ldsD[rg][cg*4+2] = accD.z; ldsD[rg][cg*4+3] = accD.w;
    ldsS[rg][cg*4+0] = accS.x; ldsS[rg][cg*4+1] = accS.y;
    ldsS[rg][cg*4+2] = accS.z; ldsS[rg][cg*4+3] = accS.w;
    ldsC[t] = cnt;
    __syncthreads();

    if (t < 128) {
        float sD = 0.f, sS = 0.f;
        #pragma unroll
        for (int g = 0; g < 8; ++g) { sD += ldsD[g][t]; sS += ldsS[g][t]; }
        wsDm[(s * B_ + b) * D_ + t] = sD;
        wsSd[(s * B_ + b) * D_ + t] = sS;
    }
    if (t == 0) {
        float c2 = 0.f;
        for (int k = 0; k < 256; ++k) c2 += ldsC[k];
        wsCnt[s * B_ + b] = c2;
        if (s == 0) acc[b] = 0.f;    // zero the per-batch accumulator for k2
    }
}

// ---------------------------------------------------------------------------
// Kernel 2: fused GEMM M = relu(dm @ W1 + b1) (never materialized) and
//   acc[b] += sum_{p,q} M[b][p,q] * (Dmd[b][p]*w2a[q] + Sd[b][p]*w2b[q])
// WMMA f32 16x16x4: tiles of 16 batches x 16 W1-columns, K = 128 (32 steps).
// Grid: (16 col-blocks of 1024 cols) x (16 batch-tiles); 256 thr = 8 waves,
// each wave owns 128 columns = 8 tiles.
// ---------------------------------------------------------------------------
__global__ __launch_bounds__(256) void k2_wmma(
    const float* __restrict__ d,  const float* __restrict__ W1,
    const float* __restrict__ b1, const float* __restrict__ W2,
    const float* __restrict__ wsDm, const float* __restrict__ wsSd,
    float* __restrict__ acc)
{
    const int cb = blockIdx.x;           // 0..15 -> 1024 columns each
    const int bt = blockIdx.y;           // 0..15 -> 16 batches each
    const int b0 = bt * 16;
    const int colBase = cb * 1024;
    const int t = threadIdx.x;
    const int w = t >> 5;                // wave id 0..7
    const int L = t & 31;                // lane

    __shared__ float As[16 * 130];       // dm tile, padded stride
    __shared__ float Dmd[16 * 128];
    __shared__ float Sdl[16 * 128];
    __shared__ float w2s[256];
    __shared__ float accLds[16];

    // Stage A tile: dm[b0+m][k] = d[(b0+m)*N*D + k]
    #pragma unroll
    for (int i = 0; i < 8; ++i) {
        const int idx = i * 256 + t;
        const int m = idx >> 7, k = idx & 127;
        As[m * 130 + k] = d[(long)(b0 + m) * N_ * D_ + k];
    }
    // Stage reduced coefficient vectors (sum the 4 N-split slabs here)
    #pragma unroll
    for (int i = 0; i < 8; ++i) {
        const int idx = i * 256 + t;     // == m*128 + p
        const int m = idx >> 7, p = idx & 127;
        float sD = 0.f, sS = 0.f;
        #pragma unroll
        for (int s2 = 0; s2 < NSPLIT; ++s2) {
            sD += wsDm[(s2 * B_ + b0 + m) * D_ + p];
            sS += wsSd[(s2 * B_ + b0 + m) * D_ + p];
        }
        Dmd[idx] = sD; Sdl[idx] = sS;
    }
    w2s[t] = W2[t];
    if (t < 16) accLds[t] = 0.f;
    __syncthreads();

    // A operand layout (f32 16x4): lanes 0-15 -> M=lane, K rows k0+0,k0+1;
    //                              lanes16-31 -> M=lane-16, K rows k0+2,k0+3.
    // B operand layout (f32 4x16): lanes 0-15 -> N=lane, K rows k0+0,k0+1;
    //                              lanes16-31 -> N=lane-16, K rows k0+2,k0+3.
    const int mrow  = L & 15;
    const int koff  = (L >> 4) * 2;
    const int mBase = (L >> 4) * 8;      // C layout: VGPR r holds M = mBase + r

    float wacc[8];
    #pragma unroll
    for (int r = 0; r < 8; ++r) wacc[r] = 0.f;

    for (int tt = 0; tt < 8; ++tt) {
        const int j0 = colBase + w * 128 + tt * 16;
        const int j  = j0 + (L & 15);    // this lane's W1 column (N index)
        v8f c = {};
        #pragma unroll 4
        for (int k0 = 0; k0 < D_; k0 += 4) {
            v2f a = *(const v2f*)&As[mrow * 130 + k0 + koff];   // 8B LDS read
            const float* wp = W1 + (long)(k0 + koff) * ND_ + j;
            v2f bb;
            bb.x = wp[0];
            bb.y = wp[ND_];
            c = __builtin_amdgcn_wmma_f32_16x16x4_f32(
                    /*neg_a=*/false, a, /*neg_b=*/false, bb,
                    /*c_mod=*/(short)0, c, /*reuse_a=*/false, /*reuse_b=*/false);
        }
        // Bias + relu + coefficient contraction (j, p, q uniform over r)
        const float bias = b1[j];
        const int q = j & 127, p = j >> 7;
        const float wa = w2s[q], wb = w2s[128 + q];
        #pragma unroll
        for (int r = 0; r < 8; ++r) {
            const int m = mBase + r;
            float val = c[r] + bias;
            val = val > 0.f ? val : 0.f;
            wacc[r] += val * (Dmd[m * 128 + p] * wa + Sdl[m * 128 + p] * wb);
        }
    }

    #pragma unroll
    for (int r = 0; r < 8; ++r) atomicAdd(&accLds[mBase + r], wacc[r]);
    __syncthreads();
    if (t < 16) atomicAdd(&acc[b0 + t], accLds[t]);
}

// ---------------------------------------------------------------------------
// Kernel 3: out[b] = sigmoid(acc[b] + b2 * cnt_diff[b])
// ---------------------------------------------------------------------------
__global__ __launch_bounds__(256) void k3_final(
    const float* __restrict__ acc, const float* __restrict__ wsCnt,
    const float* __restrict__ b2, float* __restrict__ out)
{
    const int b = threadIdx.x;
    float cnt = 0.f;
    #pragma unroll
    for (int s = 0; s < NSPLIT; ++s) cnt += wsCnt[s * B_ + b];
    const float val = acc[b] + b2[0] * cnt;
    out[b] = 1.f / (1.f + __expf(-val));
}

// ---------------------------------------------------------------------------
extern "C" void kernel_launch(void* const* d_in, const int* in_sizes, int n_in,
                              void* d_out, int out_size, void* d_ws, size_t ws_size,
                              hipStream_t stream) {
    const float*         d   = (const float*)d_in[0];
    const float*         si  = (const float*)d_in[1];
    const float*         sj  = (const float*)d_in[2];
    const unsigned char* mi  = (const unsigned char*)d_in[3];  // jax bool = 1 byte
    const unsigned char* mj  = (const unsigned char*)d_in[4];
    const float*         W1  = (const float*)d_in[5];
    const float*         b1  = (const float*)d_in[6];
    const float*         W2  = (const float*)d_in[7];
    const float*         b2  = (const float*)d_in[8];
    float* out = (float*)d_out;

    float* ws    = (float*)d_ws;
    float* wsDm  = ws;                               // NSPLIT*B*D
    float* wsSd  = wsDm + NSPLIT * B_ * D_;          // NSPLIT*B*D
    float* wsCnt = wsSd + NSPLIT * B_ * D_;          // NSPLIT*B
    float* acc   = wsCnt + NSPLIT * B_;              // B

    k1_reduce<<<B_ * NSPLIT, 256, 0, stream>>>(d, si, sj, mi, mj, wsDm, wsSd, wsCnt, acc);
    k2_wmma<<<dim3(16, 16), 256, 0, stream>>>(d, W1, b1, W2, wsDm, wsSd, acc);
    k3_final<<<1, 256, 0, stream>>>(acc, wsCnt, b2, out);
}